// LargestRectangleUtil_5531917877520
// MI455X (gfx1250) — compile-verified
//
#include <hip/hip_runtime.h>
#include <stdint.h>

#define Hh 128
#define Ww 128
#define Bb 32

typedef uint32_t u32;
typedef uint64_t u64;
typedef __attribute__((ext_vector_type(4))) u32 u32x4;
typedef __attribute__((ext_vector_type(8))) u32 u32x8;

// Low 32 bits of a generic LDS address are the byte offset within the LDS
// allocation (see CDNA5 ISA).
__device__ __forceinline__ u32 lds_off(const void* p) {
    return (u32)(uintptr_t)p;
}

__global__ __launch_bounds__(128) void init_kernel(int* out) {
    if (threadIdx.x < Bb) out[threadIdx.x] = 0;
}

// One block per image. TDM-loads the 128x128 f32 tile into LDS, runs the
// sequential column height scan, async-stores u8 heights back to scratch.
__global__ __launch_bounds__(128) void heights_kernel(const float* __restrict__ Y,
                                                      unsigned char* __restrict__ hts) {
    __shared__ float          tile[Hh * Ww];   // 64 KB
    __shared__ unsigned char  hbuf[Hh * Ww];   // 16 KB

    const int b = blockIdx.x;
    const int j = threadIdx.x;                 // column 0..127

    // ---- TDM: single tensor_load_to_lds moves the whole 64KB tile ----
    if (j < 32) {  // wave-uniform branch: only wave 0 issues the DMA
        const u64 ga = (u64)(uintptr_t)(Y + (size_t)b * Hh * Ww);
        u32x4 g0;
        g0.x = 1u;                                        // count=1, user mode
        g0.y = lds_off(&tile[0]);                         // lds_addr
        g0.z = (u32)ga;                                   // global_addr[31:0]
        g0.w = (u32)((ga >> 32) & 0x01FFFFFFu) | (2u << 30); // addr[56:32] | type=2
        u32x8 g1;
        g1.s0 = 2u << 16;          // workgroup_mask=0, data_size=2 (4 bytes)
        g1.s1 = (u32)Ww << 16;     // tensor_dim0[15:0]=128 (atomic_barrier_addr=0)
        g1.s2 = (u32)Hh << 16;     // tensor_dim0 hi=0 | tensor_dim1[15:0]=128
        g1.s3 = (u32)Ww << 16;     // tensor_dim1 hi=0 | tile_dim0=128
        g1.s4 = (u32)Hh;           // tile_dim1=128, tile_dim2=0
        g1.s5 = (u32)Ww;           // tensor_dim0_stride=128 (lo32)
        g1.s6 = 0u;                // stride hi, dim1_stride lo
        g1.s7 = 0u;                // dim1_stride hi
        asm volatile("tensor_load_to_lds %0, %1" :: "s"(g0), "s"(g1) : "memory");
        __builtin_amdgcn_s_wait_tensorcnt(0);
    }
    __syncthreads();

    // ---- sequential height recurrence down the rows (thread = column) ----
    int h = 0;
    for (int i = 0; i < Hh; ++i) {
        h = (tile[i * Ww + j] > 0.5f) ? (h + 1) : 0;
        hbuf[i * Ww + j] = (unsigned char)h;
    }
    __syncthreads();

    // ---- async store LDS -> global scratch (16 KB = 8 x 128 lanes x 16B) ----
    {
        const u64 gbase = (u64)(uintptr_t)(hts + (size_t)b * Hh * Ww);
        const u32 base  = lds_off(&hbuf[0]);
        for (int it = 0; it < 8; ++it) {
            u32 off = (u32)(it * 2048 + j * 16);
            u32 l   = base + off;
            asm volatile("global_store_async_from_lds_b128 %0, %1, %2"
                         :: "v"(off), "v"(l), "s"(gbase) : "memory");
        }
        asm volatile("s_wait_asynccnt 0" ::: "memory");
    }
}

// One block per (row, image): nearest-smaller-element scans + max reduction.
__global__ __launch_bounds__(128) void area_kernel(const unsigned char* __restrict__ hts,
                                                   int* __restrict__ out) {
    __shared__ unsigned char hrow[Ww];
    __shared__ int           wred[4];

    const int i = blockIdx.x;   // row
    const int b = blockIdx.y;   // image
    const int j = threadIdx.x;  // column

    // async-load the 128-byte height row into LDS (wave 0: 32 lanes x b32)
    if (j < 32) {
        const u64 gbase = (u64)(uintptr_t)(hts + ((size_t)b * Hh + i) * Ww);
        u32 off = (u32)j * 4u;
        u32 l   = lds_off(&hrow[0]) + off;
        asm volatile("global_load_async_to_lds_b32 %0, %1, %2"
                     :: "v"(l), "v"(off), "s"(gbase) : "memory");
        asm volatile("s_wait_asynccnt 0" ::: "memory");
    }
    __syncthreads();

    const int hj = (int)hrow[j];
    int l = j - 1;
    while (l >= 0 && (int)hrow[l] >= hj) --l;   // stops at first h[k] < h[j]
    int r = j + 1;
    while (r < Ww && (int)hrow[r] >= hj) ++r;   // stops at first h[k] < h[j]
    int area = hj * (r - l - 1);

    // wave32 max reduction
    #pragma unroll
    for (int off = 16; off > 0; off >>= 1) {
        int o = __shfl_xor(area, off, 32);
        area = area > o ? area : o;
    }
    if ((j & 31) == 0) wred[j >> 5] = area;
    __syncthreads();
    if (j == 0) {
        int m = wred[0];
        m = m > wred[1] ? m : wred[1];
        m = m > wred[2] ? m : wred[2];
        m = m > wred[3] ? m : wred[3];
        atomicMax(&out[b], m);
    }
}

extern "C" void kernel_launch(void* const* d_in, const int* in_sizes, int n_in,
                              void* d_out, int out_size, void* d_ws, size_t ws_size,
                              hipStream_t stream) {
    const float*   Y   = (const float*)d_in[0];
    int*           out = (int*)d_out;                 // reference output is int32 (32,1)
    unsigned char* hts = (unsigned char*)d_ws;        // 512 KB of scratch used

    init_kernel<<<1, 128, 0, stream>>>(out);
    heights_kernel<<<Bb, 128, 0, stream>>>(Y, hts);
    area_kernel<<<dim3(Hh, Bb), 128, 0, stream>>>(hts, out);
}